// RangeLoss_24378234372365
// MI455X (gfx1250) — compile-verified
//
#include <hip/hip_runtime.h>
#include <hip/hip_bf16.h>

typedef __attribute__((ext_vector_type(2))) float v2f;
typedef __attribute__((ext_vector_type(8))) float v8f;

#define CONF_HI 1.1f
#define CONF_LO 0.9f

__device__ __forceinline__ float loss_term(float p, float t) {
    // mask => target replaced by pred => loss 0; else (p-t)^2
    bool m = (p * CONF_HI > t) && (p * CONF_LO < t);
    float d = p - t;
    return m ? 0.0f : d * d;
}

// Wave32 all-lane sum using V_WMMA_F32_16X16X4_F32 with a ones B-matrix.
// A (16x4 f32): VGPR0 holds K=0 (lanes 0-15) / K=2 (lanes 16-31); VGPR1 -> K=1/K=3.
// With acc in A-slot0 and 0 in A-slot1: rowsum[m] = acc[m] + acc[m+16].
// D[m][n] = rowsum[m]; per-lane sum of the 8 C/D VGPRs gives
//   lanes 0-15 : sum(acc[0..7])  + sum(acc[16..23])
//   lanes16-31 : sum(acc[8..15]) + sum(acc[24..31])
// so one xor-16 shuffle-add completes the full wave sum.
__device__ __forceinline__ float wave32_sum_wmma(float acc) {
    v2f a; a[0] = acc;  a[1] = 0.0f;
    v2f b; b[0] = 1.0f; b[1] = 1.0f;
    v8f c = {};
    c = __builtin_amdgcn_wmma_f32_16x16x4_f32(
        /*neg_a=*/false, a, /*neg_b=*/false, b,
        /*c_mod=*/(short)0, c, /*reuse_a=*/false, /*reuse_b=*/false);
    float s = ((c[0] + c[1]) + (c[2] + c[3])) + ((c[4] + c[5]) + (c[6] + c[7]));
    s += __shfl_xor(s, 16, 32);
    return s;
}

__global__ void range_loss_partial_kernel(const float* __restrict__ preds,
                                          const float* __restrict__ target,
                                          float* __restrict__ partials,
                                          long long n) {
    const size_t tid    = (size_t)blockIdx.x * blockDim.x + threadIdx.x;
    const size_t stride = (size_t)gridDim.x * blockDim.x;
    const size_t n4     = (size_t)n >> 2;

    const float4* __restrict__ p4 = (const float4*)preds;
    const float4* __restrict__ t4 = (const float4*)target;

    // 4 independent accumulators to break dependency chains; bandwidth-bound
    // kernel: 128-bit loads are the thing that matters on MI455X (23.3 TB/s).
    float a0 = 0.0f, a1 = 0.0f, a2 = 0.0f, a3 = 0.0f;
    for (size_t i = tid; i < n4; i += stride) {
        float4 p = p4[i];
        float4 t = t4[i];
        a0 += loss_term(p.x, t.x);
        a1 += loss_term(p.y, t.y);
        a2 += loss_term(p.z, t.z);
        a3 += loss_term(p.w, t.w);
    }
    // scalar tail (N % 4), harmless when N is a multiple of 4
    size_t rem  = (size_t)n & 3;
    size_t base = n4 << 2;
    if (tid < rem) a0 += loss_term(preds[base + tid], target[base + tid]);

    float acc = (a0 + a1) + (a2 + a3);

    // wave32 reduction through the CDNA5 matrix pipe (EXEC all-ones here:
    // no divergence at this point in the kernel)
    float wsum = wave32_sum_wmma(acc);

    __shared__ float warp_partials[32];  // up to 1024 threads / 32
    const int lane = threadIdx.x & 31;
    const int wave = threadIdx.x >> 5;
    if (lane == 0) warp_partials[wave] = wsum;
    __syncthreads();

    if (threadIdx.x == 0) {
        const int nwaves = blockDim.x >> 5;
        float tot = 0.0f;
        for (int w = 0; w < nwaves; ++w) tot += warp_partials[w];
        partials[blockIdx.x] = tot;
    }
}

__global__ void range_loss_final_kernel(const float* __restrict__ partials,
                                        int n_part,
                                        float* __restrict__ out,
                                        float inv_n) {
    __shared__ float sm[256];
    float s = 0.0f;
    for (int i = threadIdx.x; i < n_part; i += blockDim.x) s += partials[i];
    sm[threadIdx.x] = s;
    __syncthreads();
    for (int off = 128; off > 0; off >>= 1) {
        if ((int)threadIdx.x < off) sm[threadIdx.x] += sm[threadIdx.x + off];
        __syncthreads();
    }
    if (threadIdx.x == 0) out[0] = sm[0] * inv_n;
}

extern "C" void kernel_launch(void* const* d_in, const int* in_sizes, int n_in,
                              void* d_out, int out_size, void* d_ws, size_t ws_size,
                              hipStream_t stream) {
    const float* preds  = (const float*)d_in[0];
    const float* target = (const float*)d_in[1];
    float* out = (float*)d_out;
    float* partials = (float*)d_ws;

    long long n = (long long)in_sizes[0];  // N*D flat (33,554,432)

    int threads = 256;
    int blocks  = 2048;  // 64 elems/thread at N=2^25; plenty of waves per WGP
    // never exceed scratch capacity for the per-block partials
    size_t max_blocks = ws_size / sizeof(float);
    if ((size_t)blocks > max_blocks) blocks = (int)max_blocks;
    if (blocks < 1) blocks = 1;

    float inv_n = 1.0f / (float)n;  // N = 2^25 -> exact

    range_loss_partial_kernel<<<blocks, threads, 0, stream>>>(preds, target, partials, n);
    range_loss_final_kernel<<<1, 256, 0, stream>>>(partials, blocks, out, inv_n);
}